// ESN_2D_52364241273016
// MI455X (gfx1250) — compile-verified
//
#include <hip/hip_runtime.h>

typedef __attribute__((ext_vector_type(2))) float v2f;
typedef __attribute__((ext_vector_type(8))) float v8f;

#define N_RES   128
#define BATCH   256
#define HEIGHT  32
#define WIDTH   32
#define DREAL   257          // 2n+1
#define DP      272          // padded to 17 WMMA tiles
#define LPOS    961          // 31*31 target positions
#define ALPHA2  25.0f
#define KC      64           // K-chunk (positions) staged in LDS for HtH
#define APAD    132          // padded LDS row stride (floats) -> conflict-free

static __device__ __forceinline__ v8f wmma_f32(v2f a, v2f b, v8f c) {
  // D = A(16x4 f32) * B(4x16 f32) + C(16x16 f32)
  return __builtin_amdgcn_wmma_f32_16x16x4_f32(
      /*neg_a=*/false, a, /*neg_b=*/false, b,
      /*c_mod=*/(short)0, c, /*reuse_a=*/false, /*reuse_b=*/false);
}

// ---------------------------------------------------------------------------
// Phase 1: one anti-diagonal of the 2D ESN recurrence.
// S[h][w] (layout [cell][batch][n]) = tanh(x*win + left@W1 + up@W2)
// Grid: (num_cells_on_diag * 16) blocks of 256 threads.
//   blockIdx.x/16 -> cell on diagonal, blockIdx.x%16 -> 16-row batch tile
//   wave (threadIdx.x/32) -> one of 8 column tiles of N=128
// A-slabs (16 batch rows x 128 features) are shared by all 8 waves -> stage
// once in LDS (coalesced), read back as conflict-free ds_load_b64.
// ---------------------------------------------------------------------------
__global__ void esn_cell_diag(const float* __restrict__ img,
                              const float* __restrict__ win,
                              const float* __restrict__ W1,
                              const float* __restrict__ W2,
                              float* __restrict__ S,
                              int diag) {
  __shared__ float Al[16][APAD];   // left-neighbor slab
  __shared__ float Au[16][APAD];   // up-neighbor slab

  const int h0   = (diag > (WIDTH - 1)) ? diag - (WIDTH - 1) : 0;
  const int cell = blockIdx.x >> 4;
  const int mt   = blockIdx.x & 15;
  const int h    = h0 + cell;
  const int w    = diag - h;

  const int wave  = threadIdx.x >> 5;   // N tile 0..7
  const int lane  = threadIdx.x & 31;
  const int l15   = lane & 15;
  const int khalf = (lane >> 4) << 1;   // 0 or 2 (K half per A/B layout)
  const int n     = wave * 16 + l15;    // output col this lane handles

  // Cooperative, coalesced staging of the shared A-slabs.
  if (w > 0) {
    const float* SL = S + ((size_t)(h * WIDTH + (w - 1)) * BATCH + mt * 16) * N_RES;
    for (int idx = threadIdx.x; idx < 16 * N_RES; idx += 256)
      Al[idx >> 7][idx & 127] = SL[(size_t)(idx >> 7) * N_RES + (idx & 127)];
  }
  if (h > 0) {
    const float* SU = S + ((size_t)((h - 1) * WIDTH + w) * BATCH + mt * 16) * N_RES;
    for (int idx = threadIdx.x; idx < 16 * N_RES; idx += 256)
      Au[idx >> 7][idx & 127] = SU[(size_t)(idx >> 7) * N_RES + (idx & 127)];
  }
  __syncthreads();

  v8f acc = {};
  if (w > 0) {  // left @ W1
    for (int k0 = 0; k0 < N_RES; k0 += 4) {
      v2f a = *(const v2f*)&Al[l15][k0 + khalf];
      v2f b;
      b.x = W1[(k0 + khalf) * N_RES + n];
      b.y = W1[(k0 + khalf + 1) * N_RES + n];
      acc = wmma_f32(a, b, acc);
    }
  }
  if (h > 0) {  // up @ W2
    for (int k0 = 0; k0 < N_RES; k0 += 4) {
      v2f a = *(const v2f*)&Au[l15][k0 + khalf];
      v2f b;
      b.x = W2[(k0 + khalf) * N_RES + n];
      b.y = W2[(k0 + khalf + 1) * N_RES + n];
      acc = wmma_f32(a, b, acc);
    }
  }

  // Epilogue: bias outer(x, win) + tanh, store state.
  float* SC = S + (size_t)((h * WIDTH + w) * BATCH) * N_RES;
  const int mbase = mt * 16 + ((lane >> 4) << 3);  // +8 for upper lane half
  const float wv = win[n];
#pragma unroll
  for (int r = 0; r < 8; ++r) {
    const int rowM = mbase + r;                    // batch index
    const float xv = img[(rowM * HEIGHT + h) * WIDTH + w];
    SC[(size_t)rowM * N_RES + n] = tanhf(acc[r] + xv * wv);
  }
}

// ---------------------------------------------------------------------------
// aug value: augmented design row for position l (0..960), feature d (0..271)
//   d <  128 : left neighbor state  S[h][t-1]
//   d <  256 : up neighbor state    S[h-1][t]
//   d == 256 : constant 1
//   else     : zero padding (also zero for l >= 961)
// ---------------------------------------------------------------------------
static __device__ __forceinline__ float aug_val(const float* __restrict__ S,
                                                int b, int l, int d) {
  if (l >= LPOS) return 0.0f;
  const int h = l / 31 + 1;
  const int t = l - (h - 1) * 31 + 1;
  if (d < N_RES)
    return S[((size_t)(h * WIDTH + (t - 1)) * BATCH + b) * N_RES + d];
  if (d < 2 * N_RES)
    return S[((size_t)((h - 1) * WIDTH + t) * BATCH + b) * N_RES + (d - N_RES)];
  if (d == 2 * N_RES) return 1.0f;
  return 0.0f;
}

// ---------------------------------------------------------------------------
// Phase 2: HtH[b] = Aug_b^T Aug_b + alpha^2*I (padded to 272x272).
// Grid: (3, 17, BATCH) x 256 threads. blockIdx.y = ti (shared by all 8 waves
// in the block), tj = blockIdx.x*8 + wave (active if < 17).
// Per 64-position K-chunk, the shared A-panel (64 x 16 features) is staged in
// LDS by all threads; each active wave then runs 16 WMMA steps, gathering its
// B-tile (coalesced 64B per half-wave) from global.
// K padded 961 -> 1024 with zero contributions.
// ---------------------------------------------------------------------------
__global__ void esn_hth(const float* __restrict__ S,
                        float* __restrict__ HtH) {
  __shared__ float Apan[KC][16];

  const int b    = blockIdx.z;
  const int ti   = blockIdx.y;
  const int wave = threadIdx.x >> 5;
  const int lane = threadIdx.x & 31;
  const int tj   = blockIdx.x * 8 + wave;
  const bool active = (tj < 17);              // uniform per wave
  const int l15 = lane & 15;
  const int kb  = (lane >> 4) << 1;           // 0 or 2
  const int dc  = tj * 16 + l15;              // B col (feature)

  v8f acc = {};
  for (int l0 = 0; l0 < 1024; l0 += KC) {
    __syncthreads();                          // previous panel fully consumed
    for (int idx = threadIdx.x; idx < KC * 16; idx += 256) {
      const int k = idx >> 4, f = idx & 15;
      Apan[k][f] = aug_val(S, b, l0 + k, ti * 16 + f);
    }
    __syncthreads();
    if (active) {
      for (int kk = 0; kk < KC; kk += 4) {
        v2f a, bb;
        a.x  = Apan[kk + kb][l15];
        a.y  = Apan[kk + kb + 1][l15];
        bb.x = aug_val(S, b, l0 + kk + kb,     dc);
        bb.y = aug_val(S, b, l0 + kk + kb + 1, dc);
        acc = wmma_f32(a, bb, acc);
      }
    }
  }

  if (active) {
    const int rbase = ti * 16 + ((lane >> 4) << 3);
#pragma unroll
    for (int r = 0; r < 8; ++r) {
      const int row = rbase + r;
      const int col = tj * 16 + l15;
      float v = acc[r];
      if (row == col) v += (row < DREAL) ? ALPHA2 : 1.0f;  // ridge / pad identity
      HtH[((size_t)b * DP + row) * DP + col] = v;
    }
  }
}

// ---------------------------------------------------------------------------
// Phase 2b: HtU[b][d] = sum_l aug(b,l,d) * u(b,l)  (plain reduction)
// ---------------------------------------------------------------------------
__global__ void esn_htu(const float* __restrict__ S,
                        const float* __restrict__ img,
                        float* __restrict__ HtU) {
  const int b = blockIdx.x;
  const int d = threadIdx.x;
  if (d >= DP) return;
  float s = 0.0f;
  for (int l = 0; l < LPOS; ++l) {
    const int h = l / 31 + 1;
    const int t = l - (h - 1) * 31 + 1;
    const float u = img[(b * HEIGHT + h) * WIDTH + t];
    s += aug_val(S, b, l, d) * u;
  }
  HtU[(size_t)b * DP + d] = s;
}

// ---------------------------------------------------------------------------
// Phase 3: per-batch Gauss-Jordan solve (SPD + 25*I -> no pivoting needed).
// One workgroup per batch item; rows striped over threads.
// ---------------------------------------------------------------------------
__global__ void esn_solve(float* __restrict__ HtH,
                          float* __restrict__ HtU,
                          float* __restrict__ out) {
  const int b = blockIdx.x;
  float* A   = HtH + (size_t)b * DP * DP;
  float* rhs = HtU + (size_t)b * DP;

  for (int k = 0; k < DP; ++k) {
    __threadfence();
    __syncthreads();
    const float pinv = 1.0f / A[(size_t)k * DP + k];
    for (int i = threadIdx.x; i < DP; i += blockDim.x) {
      if (i == k) continue;
      const float f = A[(size_t)i * DP + k] * pinv;
      if (f != 0.0f) {
        for (int j = k; j < DP; ++j)
          A[(size_t)i * DP + j] -= f * A[(size_t)k * DP + j];
        rhs[i] -= f * rhs[k];
      }
    }
  }
  __threadfence();
  __syncthreads();
  for (int i = threadIdx.x; i < DREAL; i += blockDim.x)
    out[(size_t)b * DREAL + i] = rhs[i] / A[(size_t)i * DP + i];
}

// ---------------------------------------------------------------------------
extern "C" void kernel_launch(void* const* d_in, const int* in_sizes, int n_in,
                              void* d_out, int out_size, void* d_ws, size_t ws_size,
                              hipStream_t stream) {
  const float* img = (const float*)d_in[0];   // [256][32][32]
  const float* win = (const float*)d_in[1];   // [128]
  const float* W1  = (const float*)d_in[2];   // [128][128]
  const float* W2  = (const float*)d_in[3];   // [128][128]
  float* out = (float*)d_out;                 // [256][257]

  char* ws = (char*)d_ws;
  const size_t S_BYTES   = (size_t)HEIGHT * WIDTH * BATCH * N_RES * sizeof(float); // 128 MB
  const size_t HTH_BYTES = (size_t)BATCH * DP * DP * sizeof(float);                // ~72 MB
  float* S   = (float*)ws;
  float* HtH = (float*)(ws + S_BYTES);
  float* HtU = (float*)(ws + S_BYTES + HTH_BYTES);

  // Phase 1: wavefront over 63 anti-diagonals.
  for (int d = 0; d < HEIGHT + WIDTH - 1; ++d) {
    const int h0 = (d > WIDTH - 1) ? d - (WIDTH - 1) : 0;
    const int h1 = (d < HEIGHT - 1) ? d : HEIGHT - 1;
    const int nc = h1 - h0 + 1;
    esn_cell_diag<<<dim3(nc * 16), dim3(256), 0, stream>>>(img, win, W1, W2, S, d);
  }
  // Phase 2: Gram matrix + ridge diagonal, then HtU.
  esn_hth<<<dim3(3, 17, BATCH), dim3(256), 0, stream>>>(S, HtH);
  esn_htu<<<dim3(BATCH), dim3(288), 0, stream>>>(S, img, HtU);
  // Phase 3: batched solve.
  esn_solve<<<dim3(BATCH), dim3(256), 0, stream>>>(HtH, HtU, out);
}